// Model_42451456753944
// MI455X (gfx1250) — compile-verified
//
#include <hip/hip_runtime.h>
#include <hip/hip_bf16.h>

#define NN   4096
#define CC   256
#define KCL  512   // cluster_num
#define KNN  32    // k nearest

typedef float v2f __attribute__((ext_vector_type(2)));
typedef float v8f __attribute__((ext_vector_type(8)));

#define FLT_BIG 3.402823466e+38f

__device__ __forceinline__ float hash01(unsigned v) {
    v ^= v >> 16; v *= 0x7feb352du;
    v ^= v >> 15; v *= 0x846ca68bu;
    v ^= v >> 16;
    return (v >> 8) * (1.0f / 16777216.0f);
}

// ---------------------------------------------------------------------------
// 1) x2[b*N+i] = sum_c x[b,i,c]^2   (one wave per row), also zero d2max bits
// ---------------------------------------------------------------------------
__global__ void rowsumsq_kernel(const float* __restrict__ X, float* __restrict__ X2,
                                unsigned* __restrict__ d2maxbits) {
    int lane = threadIdx.x & 31;
    int row  = blockIdx.x * 8 + (threadIdx.x >> 5);       // 0 .. B*N-1
    const float* p = X + (size_t)row * CC;
    float s = 0.0f;
    #pragma unroll
    for (int t = 0; t < CC / 32; ++t) {
        float v = p[lane + t * 32];
        s = fmaf(v, v, s);
    }
    #pragma unroll
    for (int off = 16; off; off >>= 1) s += __shfl_xor(s, off);
    if (lane == 0) X2[row] = s;
    if (blockIdx.x == 0 && threadIdx.x < 4) d2maxbits[threadIdx.x] = 0u;
}

// ---------------------------------------------------------------------------
// 2) d2[i][j] = x2[i] + x2[j] - 2 * x_i . x_j   via V_WMMA_F32_16X16X4_F32
//    Workgroup tile 128x64 (8 waves = 4 row-bands x 2 col-bands),
//    each wave computes a 32x32 sub-tile: 4 accumulators, WMMA:ds ratio 1:1.
// ---------------------------------------------------------------------------
__global__ void __launch_bounds__(256)
gram_d2_kernel(const float* __restrict__ Xb, const float* __restrict__ X2b,
               float* __restrict__ D2) {
    __shared__ float lA[128][68];  // pad 68: float4-aligned, conflict-free frags
    __shared__ float lB[64][68];

    const int tid  = threadIdx.x;
    const int lane = tid & 31;
    const int wv   = tid >> 5;        // 0..7
    const int wr   = wv >> 1;         // 0..3  -> 32-row band
    const int wc   = wv & 1;          // 0..1  -> 32-col band
    const int i0   = blockIdx.y * 128;
    const int j0   = blockIdx.x * 64;

    const int n    = lane & 15;           // fragment row/col within 16
    const int koff = (lane >> 4) << 1;    // 0 or 2

    v8f acc[2][2];
    acc[0][0] = (v8f){}; acc[0][1] = (v8f){};
    acc[1][0] = (v8f){}; acc[1][1] = (v8f){};

    for (int kc = 0; kc < CC; kc += 64) {
        // stage A: 128 rows x 64 cols (2048 float4, 8 iters of 256 threads)
        #pragma unroll
        for (int it = 0; it < 8; ++it) {
            int idx = it * 256 + tid;          // 0..2047
            int r   = idx >> 4;                // 0..127
            int c4  = (idx & 15) << 2;         // 0..60
            float4 va = *(const float4*)&Xb[(size_t)(i0 + r) * CC + kc + c4];
            *(float4*)&lA[r][c4] = va;
        }
        // stage B: 64 rows x 64 cols (1024 float4, 4 iters)
        #pragma unroll
        for (int it = 0; it < 4; ++it) {
            int idx = it * 256 + tid;          // 0..1023
            int r   = idx >> 4;                // 0..63
            int c4  = (idx & 15) << 2;
            float4 vb = *(const float4*)&Xb[(size_t)(j0 + r) * CC + kc + c4];
            *(float4*)&lB[r][c4] = vb;
        }
        __syncthreads();

        #pragma unroll
        for (int k = 0; k < 64; k += 4) {
            v2f a0 = *(const v2f*)&lA[wr * 32 + n][k + koff];
            v2f a1 = *(const v2f*)&lA[wr * 32 + 16 + n][k + koff];
            v2f b0 = *(const v2f*)&lB[wc * 32 + n][k + koff];
            v2f b1 = *(const v2f*)&lB[wc * 32 + 16 + n][k + koff];
            acc[0][0] = __builtin_amdgcn_wmma_f32_16x16x4_f32(false, a0, false, b0,
                                                              (short)0, acc[0][0], false, false);
            acc[0][1] = __builtin_amdgcn_wmma_f32_16x16x4_f32(false, a0, false, b1,
                                                              (short)0, acc[0][1], false, false);
            acc[1][0] = __builtin_amdgcn_wmma_f32_16x16x4_f32(false, a1, false, b0,
                                                              (short)0, acc[1][0], false, false);
            acc[1][1] = __builtin_amdgcn_wmma_f32_16x16x4_f32(false, a1, false, b1,
                                                              (short)0, acc[1][1], false, false);
        }
        __syncthreads();
    }

    // epilogue: d2 = x2_i + x2_j - 2*dot
    const int rbase = i0 + wr * 32;
    const int cbase = j0 + wc * 32;
    #pragma unroll
    for (int ia = 0; ia < 2; ++ia) {
        const int mb = rbase + ia * 16 + ((lane >> 4) << 3);
        float x2i[8];
        #pragma unroll
        for (int r = 0; r < 8; ++r) x2i[r] = X2b[mb + r];
        #pragma unroll
        for (int jb = 0; jb < 2; ++jb) {
            const int jg = cbase + jb * 16 + n;
            const float x2j = X2b[jg];
            #pragma unroll
            for (int r = 0; r < 8; ++r) {
                D2[(size_t)(mb + r) * NN + jg] = x2i[r] + x2j - 2.0f * acc[ia][jb][r];
            }
        }
    }
}

// ---------------------------------------------------------------------------
// 3) density: one wave per row. Per-lane top-32-smallest of 128 d2 values
//    (insertion list), then 32-way merge across lanes for the global 32
//    smallest; density = exp(-sum/(k*C)) + noise. Also atomic-max of d2.
// ---------------------------------------------------------------------------
__global__ void __launch_bounds__(256)
density_kernel(const float* __restrict__ D2, float* __restrict__ density,
               unsigned* __restrict__ d2maxbits, int b) {
    __shared__ float cand[8][32][32];
    const int lane = threadIdx.x & 31;
    const int w    = threadIdx.x >> 5;
    const int i    = blockIdx.x * 8 + w;
    const float* row = D2 + (size_t)i * NN;

    float t32[32];
    #pragma unroll
    for (int s = 0; s < 32; ++s) t32[s] = FLT_BIG;
    float rmax = 0.0f;

    for (int t = 0; t < NN / 32; ++t) {
        float v = fmaxf(row[lane + t * 32], 0.0f);
        rmax = fmaxf(rmax, v);
        if (v < t32[31]) {
            t32[31] = v;
            #pragma unroll
            for (int s = 31; s >= 1; --s) {
                float a = t32[s - 1];
                if (t32[s] < a) { t32[s - 1] = t32[s]; t32[s] = a; }
            }
        }
    }
    #pragma unroll
    for (int s = 0; s < 32; ++s) cand[w][lane][s] = t32[s];
    __syncthreads();

    // 32-way sorted-list merge: take 32 smallest overall, accumulate sum
    int ptr = 0;
    float sum = 0.0f;
    for (int it = 0; it < KNN; ++it) {
        float mv = cand[w][lane][ptr];
        int   ml = lane;
        #pragma unroll
        for (int off = 16; off; off >>= 1) {
            float ov = __shfl_xor(mv, off);
            int   ol = __shfl_xor(ml, off);
            if (ov < mv || (ov == mv && ol < ml)) { mv = ov; ml = ol; }
        }
        sum += mv;
        if (lane == ml) ptr++;
    }
    #pragma unroll
    for (int off = 16; off; off >>= 1) rmax = fmaxf(rmax, __shfl_xor(rmax, off));

    if (lane == 0) {
        float dens = __expf(-sum * (1.0f / (KNN * CC)))
                   + hash01((unsigned)(b * NN + i) * 0x9e3779b9u + 1u) * 1e-6f;
        density[b * NN + i] = dens;
        atomicMax(d2maxbits + b, __float_as_uint(rmax));
    }
}

// ---------------------------------------------------------------------------
// 4) dist-to-higher-density + score. One 256-thread block per row.
// ---------------------------------------------------------------------------
__global__ void __launch_bounds__(256)
distmin_kernel(const float* __restrict__ D2, const float* __restrict__ density,
               const unsigned* __restrict__ d2maxbits, float* __restrict__ score, int b) {
    __shared__ float red[8];
    const int i = blockIdx.x;
    const float di = density[b * NN + i];
    float m2 = __uint_as_float(d2maxbits[b]);
    const float* row = D2 + (size_t)i * NN;
    for (int j = threadIdx.x; j < NN; j += 256) {
        if (density[b * NN + j] > di) m2 = fminf(m2, fmaxf(row[j], 0.0f));
    }
    #pragma unroll
    for (int off = 16; off; off >>= 1) m2 = fminf(m2, __shfl_xor(m2, off));
    if ((threadIdx.x & 31) == 0) red[threadIdx.x >> 5] = m2;
    __syncthreads();
    if (threadIdx.x == 0) {
        float mm = red[0];
        #pragma unroll
        for (int s = 1; s < 8; ++s) mm = fminf(mm, red[s]);
        float dist = sqrtf(mm) * (1.0f / 16.0f);   // / sqrt(C)
        score[b * NN + i] = dist * di;
    }
}

// ---------------------------------------------------------------------------
// 5) top-512 of score (single WG of 1024 threads per batch)
// ---------------------------------------------------------------------------
__global__ void __launch_bounds__(1024)
top512_kernel(const float* __restrict__ score, int* __restrict__ idxdown,
              float* __restrict__ out_index_down, int b) {
    __shared__ float sv[NN];
    __shared__ float wvv[32];
    __shared__ int   wvi[32];
    const int tid  = threadIdx.x;
    const int lane = tid & 31;
    const int wid  = tid >> 5;
    for (int j = tid; j < NN; j += 1024) sv[j] = score[b * NN + j];
    __syncthreads();

    for (int kk = 0; kk < KCL; ++kk) {
        float bv = -FLT_BIG; int bi = NN;
        #pragma unroll
        for (int t = 0; t < NN / 1024; ++t) {
            int j = tid + t * 1024;
            float v = sv[j];
            if (v > bv || (v == bv && j < bi)) { bv = v; bi = j; }
        }
        #pragma unroll
        for (int off = 16; off; off >>= 1) {
            float ov = __shfl_xor(bv, off);
            int   oi = __shfl_xor(bi, off);
            if (ov > bv || (ov == bv && oi < bi)) { bv = ov; bi = oi; }
        }
        if (lane == 0) { wvv[wid] = bv; wvi[wid] = bi; }
        __syncthreads();
        if (tid < 32) {
            bv = wvv[tid]; bi = wvi[tid];
            #pragma unroll
            for (int off = 16; off; off >>= 1) {
                float ov = __shfl_xor(bv, off);
                int   oi = __shfl_xor(bi, off);
                if (ov > bv || (ov == bv && oi < bi)) { bv = ov; bi = oi; }
            }
            if (tid == 0) {
                idxdown[b * KCL + kk] = bi;
                out_index_down[b * KCL + kk] = (float)bi;
                sv[bi] = -FLT_BIG;
            }
        }
        __syncthreads();
    }
}

// ---------------------------------------------------------------------------
// 6) assignment: argmin over the 512 center rows of d2
// ---------------------------------------------------------------------------
__global__ void __launch_bounds__(256)
assign_kernel(const float* __restrict__ D2, const int* __restrict__ idxdown,
              float* __restrict__ out_idx_cluster, int b) {
    __shared__ int cidx[KCL];
    for (int t = threadIdx.x; t < KCL; t += 256) cidx[t] = idxdown[b * KCL + t];
    __syncthreads();
    const int j = blockIdx.x * 256 + threadIdx.x;
    float best = FLT_BIG; int bk = 0;
    for (int kk = 0; kk < KCL; ++kk) {
        if (kk + 1 < KCL) {
            // gfx1250 global_prefetch_b8: pull next gathered row into cache
            __builtin_prefetch(&D2[(size_t)cidx[kk + 1] * NN + j], 0, 0);
        }
        float v = fmaxf(D2[(size_t)cidx[kk] * NN + j], 0.0f);
        if (v < best) { best = v; bk = kk; }
    }
    out_idx_cluster[b * NN + j] = (float)bk;
}

// 7) force centers to own cluster
__global__ void centerfix_kernel(const int* __restrict__ idxdown,
                                 float* __restrict__ out_idx_cluster, int b) {
    int kk = blockIdx.x * 256 + threadIdx.x;
    if (kk < KCL) out_idx_cluster[b * NN + idxdown[b * KCL + kk]] = (float)kk;
}

// 8) gather x_down = x[b, index_down, :]
__global__ void __launch_bounds__(256)
gather_kernel(const float* __restrict__ Xb, const int* __restrict__ idxdown,
              float* __restrict__ out_xdown, int b) {
    const int kk = blockIdx.x;          // 0..511
    const int c  = threadIdx.x;         // 0..255
    const int src = idxdown[b * KCL + kk];
    out_xdown[((size_t)(b * KCL + kk)) * CC + c] = Xb[(size_t)src * CC + c];
}

// ---------------------------------------------------------------------------
extern "C" void kernel_launch(void* const* d_in, const int* in_sizes, int n_in,
                              void* d_out, int out_size, void* d_ws, size_t ws_size,
                              hipStream_t stream) {
    const float* X = (const float*)d_in[0];
    const int B = 4;

    float* out            = (float*)d_out;
    float* out_index_down = out;                      // B*KCL
    float* out_idx_clust  = out + B * KCL;            // B*NN
    float* out_xdown      = out + B * KCL + B * NN;   // B*KCL*CC

    char* ws = (char*)d_ws;
    size_t off = 0;
    float* D2 = (float*)(ws + off);        off += (size_t)NN * NN * 4;   // 64 MB, reused per batch
    float* X2 = (float*)(ws + off);        off += (size_t)B * NN * 4;
    float* density = (float*)(ws + off);   off += (size_t)B * NN * 4;
    float* score = (float*)(ws + off);     off += (size_t)B * NN * 4;
    unsigned* d2maxbits = (unsigned*)(ws + off); off += 64;
    int* idxdown = (int*)(ws + off);       off += (size_t)B * KCL * 4;

    rowsumsq_kernel<<<B * NN / 8, 256, 0, stream>>>(X, X2, d2maxbits);

    for (int b = 0; b < B; ++b) {
        const float* Xb  = X  + (size_t)b * NN * CC;
        const float* X2b = X2 + (size_t)b * NN;
        dim3 gg(NN / 64, NN / 128);
        gram_d2_kernel<<<gg, 256, 0, stream>>>(Xb, X2b, D2);
        density_kernel<<<NN / 8, 256, 0, stream>>>(D2, density, d2maxbits, b);
        distmin_kernel<<<NN, 256, 0, stream>>>(D2, density, d2maxbits, score, b);
        top512_kernel<<<1, 1024, 0, stream>>>(score, idxdown, out_index_down, b);
        assign_kernel<<<NN / 256, 256, 0, stream>>>(D2, idxdown, out_idx_clust, b);
        centerfix_kernel<<<(KCL + 255) / 256, 256, 0, stream>>>(idxdown, out_idx_clust, b);
        gather_kernel<<<KCL, 256, 0, stream>>>(Xb, idxdown, out_xdown, b);
    }
}